// ParallelLSTM_52819507806514
// MI455X (gfx1250) — compile-verified
//
#include <hip/hip_runtime.h>

// ---------------------------------------------------------------------------
// ParallelLSTM for MI455X (gfx1250, wave32).
//
// Algebraic restructuring (exact up to fp32 reassociation):
//   W    = w2 @ w1                      [15,15]
//   bf   = w2 @ b1 + b2                 [15]
//   forecasts = x @ W.T + bf            [S,15]
//   Wpre = w_ih[:, :15] + w_ih[:, 15:] @ W     [4096,15]
//   bpre = w_ih[:, 15:] @ bf + b_ih + b_hh     [4096]
//   pre  = x @ Wpre.T + bpre            [S,4096]   <- single WMMA GEMM
// Sequential LSTM scan: h is scalar (proj_size=1); one 1024-thread workgroup
// owns c[1024] in registers. Gate rows of `pre` are streamed into LDS with the
// CDNA5 async-copy engine (GLOBAL_LOAD_ASYNC_TO_LDS_B128 + s_wait_asynccnt),
// 3-slot ring buffer, one s_barrier per step shared with the h-reduction.
// pre is produced/consumed in chunks to bound workspace (~33 MB at T=2048).
// ---------------------------------------------------------------------------

typedef float v2f __attribute__((ext_vector_type(2)));
typedef float v8f __attribute__((ext_vector_type(8)));
typedef int   v4i __attribute__((ext_vector_type(4)));

typedef __attribute__((address_space(1))) v4i* gptr_v4i;  // global int4*
typedef __attribute__((address_space(3))) v4i* lptr_v4i;  // LDS int4*

#define S_LEN 65536
#define HID   1024
#define G4    4096   // 4*HID
#define NBUF  3      // async LDS pipeline depth (48KB + partials < 64KB)

// workspace layout (float offsets)
#define WPAD_OFF 0       // 16*16   padded W
#define BF_OFF   256     // 16      padded bf
#define WPRE_OFF 512     // 4096*16 padded Wpre (row-major, stride 16)
#define BPRE_OFF 66048   // 4096
#define CST_OFF  70144   // 1024    carried cell state
#define HST_OFF  71168   // 1 (+pad) carried hidden scalar
#define PRE_OFF  71680   // T*4096  pre chunk (256-aligned)

#if defined(__has_builtin)
#if __has_builtin(__builtin_amdgcn_global_load_async_to_lds_b128) && \
    __has_builtin(__builtin_amdgcn_s_wait_asynccnt)
#define ASYNC_VIA_BUILTIN 1
#endif
#endif

// Copy 16 bytes global -> LDS via the async engine (tracked with ASYNCcnt).
__device__ __forceinline__ void async_copy16(const float* g, float* l) {
#if defined(ASYNC_VIA_BUILTIN)
    __builtin_amdgcn_global_load_async_to_lds_b128(
        (gptr_v4i)(uintptr_t)g,
        (lptr_v4i)(unsigned)(uintptr_t)l,   // low 32 bits = LDS byte offset
        0, 0);
#else
    unsigned loff = (unsigned)(uintptr_t)l;
    asm volatile("global_load_async_to_lds_b128 %0, %1, off"
                 :: "v"(loff), "v"(g) : "memory");
#endif
}

#if defined(ASYNC_VIA_BUILTIN)
#define WAIT_ASYNC_LE(n) __builtin_amdgcn_s_wait_asynccnt(n)
#else
#define WAIT_ASYNC_LE(n) asm volatile("s_wait_asynccnt %0" :: "n"(n) : "memory")
#endif

__device__ __forceinline__ float sigm_f(float x) {
    return 1.0f / (1.0f + __expf(-x));
}
__device__ __forceinline__ float tanh_f(float x) {
    float e = __expf(-2.0f * x);
    return (1.0f - e) / (1.0f + e);
}
__device__ __forceinline__ float wave_reduce_add(float v) {
    #pragma unroll
    for (int off = 16; off > 0; off >>= 1) v += __shfl_xor(v, off, 32);
    return v;
}

// --- prep 1: W = w2@w1 (zero-padded 16x16), bf = w2@b1 + b2 (padded 16) ----
__global__ __launch_bounds__(256) void prep_head(
    const float* __restrict__ w1, const float* __restrict__ b1,
    const float* __restrict__ w2, const float* __restrict__ b2,
    float* __restrict__ Wpad, float* __restrict__ bf)
{
    int tid = threadIdx.x;          // 256 threads: (i,k) in 16x16
    int i = tid >> 4, k = tid & 15;
    if (i < 15 && k < 15) {
        float acc = 0.0f;
        for (int f = 0; f < 512; ++f) acc += w2[i * 512 + f] * w1[f * 15 + k];
        Wpad[i * 16 + k] = acc;
    } else {
        Wpad[i * 16 + k] = 0.0f;
    }
    if (k == 15) {
        if (i < 15) {
            float acc = b2[i];
            for (int f = 0; f < 512; ++f) acc += w2[i * 512 + f] * b1[f];
            bf[i] = acc;
        } else {
            bf[15] = 0.0f;
        }
    }
}

// --- prep 2: Wpre[4096][16] (k=15 zero), bpre[4096] ------------------------
__global__ __launch_bounds__(256) void prep_gates(
    const float* __restrict__ w_ih, const float* __restrict__ b_ih,
    const float* __restrict__ b_hh, const float* __restrict__ Wpad,
    const float* __restrict__ bf, float* __restrict__ Wpre,
    float* __restrict__ bpre)
{
    int n = blockIdx.x * 256 + threadIdx.x;   // 0..4095
    float row[16];
    #pragma unroll
    for (int k = 0; k < 15; ++k) row[k] = w_ih[n * 30 + k];
    row[15] = 0.0f;
    float bb = b_ih[n] + b_hh[n];
    #pragma unroll
    for (int j = 0; j < 15; ++j) {
        float wfj = w_ih[n * 30 + 15 + j];
        bb += wfj * bf[j];
        #pragma unroll
        for (int k = 0; k < 15; ++k) row[k] += wfj * Wpad[j * 16 + k];
    }
    #pragma unroll
    for (int k = 0; k < 16; ++k) Wpre[n * 16 + k] = row[k];
    bpre[n] = bb;
}

// --- forecasts + zero forecasted_progress ----------------------------------
__global__ __launch_bounds__(256) void forecast_out(
    const float* __restrict__ x, const float* __restrict__ Wpad,
    const float* __restrict__ bf, float* __restrict__ out)
{
    int s = blockIdx.x * 256 + threadIdx.x;   // 0..65535
    float xv[15];
    #pragma unroll
    for (int k = 0; k < 15; ++k) xv[k] = x[(size_t)s * 15 + k];
    out[S_LEN + s] = 0.0f;                    // forecasted_progress = 0
    #pragma unroll
    for (int i = 0; i < 15; ++i) {
        float o = bf[i];
        #pragma unroll
        for (int k = 0; k < 15; ++k) o += Wpad[i * 16 + k] * xv[k];
        out[(size_t)2 * S_LEN + (size_t)s * 15 + i] = o;
    }
}

// --- main GEMM: pre[T,4096] = x[rows,15] @ Wpre.T + bpre, via WMMA f32 16x16x4
// Block = 256 threads = 8 waves. Block covers one 16-row M tile and 8 N tiles
// (128 of 4096 gate columns). grid = (32, T/16).
__global__ __launch_bounds__(256) void gemm_pre(
    const float* __restrict__ x, const float* __restrict__ Wpre,
    const float* __restrict__ bpre, float* __restrict__ pre, int rowBase)
{
    __shared__ float xs[16][16];              // zero-padded x tile
    int tid = threadIdx.x;
    int mTile = blockIdx.y;
    {
        int r = tid >> 4, k = tid & 15;
        int srow = rowBase + mTile * 16 + r;
        xs[r][k] = (k < 15) ? x[(size_t)srow * 15 + k] : 0.0f;
    }
    __syncthreads();

    int lane = tid & 31, wave = tid >> 5;
    int half = lane >> 4, li = lane & 15;
    int n = blockIdx.x * 128 + wave * 16 + li;   // this lane's N column

    v8f acc = {};
    #pragma unroll
    for (int ks = 0; ks < 4; ++ks) {
        int k = ks * 4 + half * 2;
        // A frag: 16x4 f32 — lanes 0-15 hold K={0,1}, lanes 16-31 K={2,3}
        v2f a; a.x = xs[li][k];            a.y = xs[li][k + 1];
        // B frag: 4x16 f32 — B[k][n] = Wpre[n][k] (padded stride-16 rows)
        v2f b; b.x = Wpre[n * 16 + k];     b.y = Wpre[n * 16 + k + 1];
        acc = __builtin_amdgcn_wmma_f32_16x16x4_f32(
            /*neg_a=*/false, a, /*neg_b=*/false, b,
            /*c_mod=*/(short)0, acc, /*reuse_a=*/false, /*reuse_b=*/false);
    }

    float bias = bpre[n];
    int mBase = mTile * 16 + half * 8;     // D: VGPR r holds M = r + half*8
    #pragma unroll
    for (int r = 0; r < 8; ++r) {
        pre[(size_t)(mBase + r) * G4 + n] = acc[r] + bias;
    }
}

// --- sequential scan: one workgroup, c[1024] in registers, scalar h --------
// pre rows stream through a 3-slot LDS ring via async copies. Per step:
//   read slot t (ds_load) -> gate math -> wave shuffle reduce -> LDS partials
//   s_wait_asynccnt (own step-t+1 copy landed, in-order retirement)
//   s_barrier (reduction exchange + cross-wave async visibility)
//   read partials -> h ; issue async copy for step t+NBUF into freed slot t
__global__ __launch_bounds__(1024) void lstm_scan(
    const float* __restrict__ pre, const float* __restrict__ w_hh,
    const float* __restrict__ w_hr, float* __restrict__ prog,
    float* __restrict__ cState, float* __restrict__ hState, int t0, int T)
{
    __shared__ float buf[NBUF][G4];       // 48KB ring of pre rows
    __shared__ float partial[2][32];      // double-buffered wave partials

    int j = threadIdx.x;                  // channel 0..1023
    int lane = j & 31, wave = j >> 5;

    float c, h;
    if (t0 == 0) { c = 0.0f; h = 0.0f; }
    else         { c = cState[j]; h = hState[0]; }

    float wi = w_hh[j], wf = w_hh[HID + j];
    float wg = w_hh[2 * HID + j], wo = w_hh[3 * HID + j];
    float whr = w_hr[j];

    // prologue: fill the pipeline (thread j copies bytes 16j..16j+15 of a row)
    #pragma unroll
    for (int s = 0; s < NBUF; ++s) {
        if (s < T) async_copy16(pre + (size_t)s * G4 + 4 * j, &buf[s][4 * j]);
    }
    WAIT_ASYNC_LE(NBUF - 1);              // own copy of step 0 retired
    __syncthreads();                      // all waves' copies of step 0 visible

    int slot = 0;
    for (int t = 0; t < T; ++t) {
        const float* row = buf[slot];
        float gi = row[j]           + wi * h;
        float gf = row[HID + j]     + wf * h;
        float gg = row[2 * HID + j] + wg * h;
        float go = row[3 * HID + j] + wo * h;

        float iv = sigm_f(gi), fv = sigm_f(gf), ov = sigm_f(go);
        float gv = tanh_f(gg);
        c = fv * c + iv * gv;
        float r = whr * ov * tanh_f(c);

        r = wave_reduce_add(r);                    // 32 lanes -> wave sum
        if (lane == 0) partial[t & 1][wave] = r;   // 32 waves -> LDS
        WAIT_ASYNC_LE(NBUF - 2);                   // own step-t+1 copy retired
        __syncthreads();                           // one barrier per step
        float s = partial[t & 1][lane];            // every wave re-reduces
        h = wave_reduce_add(s);                    // identical order => uniform h

        int tn = t + NBUF;                         // refill freed slot (safe:
        if (tn < T)                                //  all reads of slot t done)
            async_copy16(pre + (size_t)tn * G4 + 4 * j, &buf[slot][4 * j]);

        if (j == 0) prog[t0 + t] = h;
        slot = (slot + 1 == NBUF) ? 0 : slot + 1;
    }
    cState[j] = c;
    if (j == 0) hState[0] = h;
}

extern "C" void kernel_launch(void* const* d_in, const int* in_sizes, int n_in,
                              void* d_out, int out_size, void* d_ws, size_t ws_size,
                              hipStream_t stream)
{
    const float* x    = (const float*)d_in[0];
    const float* w1   = (const float*)d_in[1];
    const float* b1   = (const float*)d_in[2];
    const float* w2   = (const float*)d_in[3];
    const float* b2   = (const float*)d_in[4];
    const float* w_ih = (const float*)d_in[5];
    const float* b_ih = (const float*)d_in[6];
    const float* w_hh = (const float*)d_in[7];
    const float* b_hh = (const float*)d_in[8];
    const float* w_hr = (const float*)d_in[9];
    float* out = (float*)d_out;
    float* ws  = (float*)d_ws;

    float* Wpad = ws + WPAD_OFF;
    float* bf   = ws + BF_OFF;
    float* Wpre = ws + WPRE_OFF;
    float* bpre = ws + BPRE_OFF;
    float* cSt  = ws + CST_OFF;
    float* hSt  = ws + HST_OFF;
    float* preC = ws + PRE_OFF;

    // chunk length: largest power-of-two (<=2048, >=16) whose pre-chunk fits ws
    int T = 2048;
    while (T > 16 &&
           ((size_t)PRE_OFF + (size_t)T * G4) * sizeof(float) > ws_size) {
        T >>= 1;
    }

    prep_head<<<1, 256, 0, stream>>>(w1, b1, w2, b2, Wpad, bf);
    prep_gates<<<G4 / 256, 256, 0, stream>>>(w_ih, b_ih, b_hh, Wpad, bf, Wpre, bpre);
    forecast_out<<<S_LEN / 256, 256, 0, stream>>>(x, Wpad, bf, out);

    for (int t0 = 0; t0 < S_LEN; t0 += T) {
        gemm_pre<<<dim3(32, T / 16), 256, 0, stream>>>(x, Wpre, bpre, preC, t0);
        lstm_scan<<<1, 1024, 0, stream>>>(preC, w_hh, w_hr, out, cSt, hSt, t0, T);
    }
}